// SSIMLoss_6622839570739
// MI455X (gfx1250) — compile-verified
//
#include <hip/hip_runtime.h>
#include <stdint.h>

// ---------------------------------------------------------------------------
// SSIM loss, B=16 C=1 T=20 H=W=256, WIN=7 -> 320 frames, 250x250 valid output
// Memory-bound: ~250MB total traffic @ 23.3TB/s -> ~11us floor. Compute is
// trivial (separable sliding box filter), so the kernel is built around the
// gfx1250 async global->LDS DMA path (ASYNCcnt) for latency hiding.
// ---------------------------------------------------------------------------

#define B_   16
#define T_   20
#define W_   256
#define HW_  65536
#define F_   320          // B_*T_
#define OUT_W 250
#define BANDS 10
#define ROWS_OUT 25       // 250 / BANDS

typedef __attribute__((address_space(1))) int as1_int;
typedef __attribute__((address_space(3))) int as3_int;

#if defined(__has_builtin)
#if __has_builtin(__builtin_amdgcn_global_load_async_to_lds_b32)
#define HAVE_ASYNC_LDS 1
#endif
#if __has_builtin(__builtin_amdgcn_s_wait_asynccnt)
#define HAVE_WAIT_ASYNC 1
#endif
#endif

__device__ __forceinline__ void stage_load(const float* gp, float* lp) {
#if defined(HAVE_ASYNC_LDS)
    // per-lane async DMA: each lane copies 4 bytes from global address gp to LDS address lp; tracked by ASYNCcnt
    __builtin_amdgcn_global_load_async_to_lds_b32(
        (as1_int*)(unsigned long long)(size_t)gp,
        (as3_int*)(unsigned int)(size_t)lp,
        0, 0);
#else
    *lp = *gp;   // fallback: synchronous copy through VGPR
#endif
}

#if defined(HAVE_ASYNC_LDS)
#if defined(HAVE_WAIT_ASYNC)
#define WAIT_ASYNC(n) __builtin_amdgcn_s_wait_asynccnt(n)
#else
#define WAIT_ASYNC(n) asm volatile("s_wait_asynccnt " #n ::: "memory")
#endif
#else
#define WAIT_ASYNC(n) ((void)0)
#endif

// ---------------------------------------------------------------------------
__global__ __launch_bounds__(32)
void k_init(float* dr, double* acc) {
    const int i = threadIdx.x;
    if (i < T_) dr[i] = 0.0f;
    if (i == 0) *acc = 0.0;
}

// per-timestep max over Yt[:, :, t, :, :]  (values are uniform[0,1) -> >= 0,
// so float ordering == int ordering of the bit patterns)
__global__ __launch_bounds__(256)
void k_framemax(const float* __restrict__ Y, float* __restrict__ dr) {
    const int t = blockIdx.x;          // 20
    const int b = blockIdx.y;          // 16
    const int chunk = blockIdx.z;      // 8
    const int tid = threadIdx.x;       // 256
    const size_t base = ((size_t)b * T_ + t) * HW_;
    const float4* p = (const float4*)(Y + base) + (size_t)chunk * 2048;

    float m = 0.0f;
    for (int i = tid; i < 2048; i += 256) {
        float4 v = p[i];
        m = fmaxf(m, fmaxf(fmaxf(v.x, v.y), fmaxf(v.z, v.w)));
    }
    __shared__ float sm[256];
    sm[tid] = m;
    __syncthreads();
    for (int s = 128; s; s >>= 1) {
        if (tid < s) sm[tid] = fmaxf(sm[tid], sm[tid + s]);
        __syncthreads();
    }
    if (tid == 0) atomicMax((int*)&dr[t], __float_as_int(sm[0]));
}

// one block per (frame, band of 25 output rows); thread w owns column w.
__global__ __launch_bounds__(256)
void k_ssim(const float* __restrict__ X, const float* __restrict__ Y,
            const float* __restrict__ dr, double* __restrict__ acc) {
    const int f    = blockIdx.x;       // 320 (frame = b*T + t)
    const int band = blockIdx.y;       // 10
    const int w    = threadIdx.x;      // 256
    const int t    = f % T_;
    const size_t base = (size_t)f * HW_;
    const float* Xf = X + base;
    const float* Yf = Y + base;
    const int r0 = band * ROWS_OUT;

    __shared__ float ringX[7][W_], ringY[7][W_];   // last 7 raw rows (lane-private slots)
    __shared__ float sVx[W_], sVy[W_], sVxx[W_], sVyy[W_], sVxy[W_];
    __shared__ float stX[4][W_], stY[4][W_];       // async stage, 3 rows in flight
    __shared__ float red[256];

    const float drt = dr[t];
    const float c1 = (0.01f * drt) * (0.01f * drt);
    const float c2 = (0.03f * drt) * (0.03f * drt);

    // prologue: DMA ring rows r0..r0+6 and stage input rows j=7,8,9
#pragma unroll
    for (int h = 0; h < 7; ++h) {
        stage_load(Xf + (size_t)(r0 + h) * W_ + w, &ringX[h][w]);
        stage_load(Yf + (size_t)(r0 + h) * W_ + w, &ringY[h][w]);
    }
#pragma unroll
    for (int j = 7; j < 10; ++j) {
        stage_load(Xf + (size_t)(r0 + j) * W_ + w, &stX[j & 3][w]);
        stage_load(Yf + (size_t)(r0 + j) * W_ + w, &stY[j & 3][w]);
    }
    WAIT_ASYNC(6);   // in-order completion: ring rows done, 3 stage rows may remain

    // initial 7-row column sums of the 5 moments (registers)
    float vx = 0.f, vy = 0.f, vxx = 0.f, vyy = 0.f, vxy = 0.f;
#pragma unroll
    for (int h = 0; h < 7; ++h) {
        float x = ringX[h][w], y = ringY[h][w];
        vx += x; vy += y; vxx += x * x; vyy += y * y; vxy += x * y;
    }

    const float inv = 1.0f / 49.0f;
    const float cn  = 49.0f / 48.0f;   // COV_NORM
    float lsum = 0.f;

    for (int hp = 0; hp < ROWS_OUT; ++hp) {
        // publish column sums for horizontal 7-tap
        sVx[w] = vx; sVy[w] = vy; sVxx[w] = vxx; sVyy[w] = vyy; sVxy[w] = vxy;

        // keep the DMA pipe 3 rows deep
        const int jiss = hp + 10;
        if (jiss <= ROWS_OUT + 5) {
            stage_load(Xf + (size_t)(r0 + jiss) * W_ + w, &stX[jiss & 3][w]);
            stage_load(Yf + (size_t)(r0 + jiss) * W_ + w, &stY[jiss & 3][w]);
        }
        __syncthreads();   // A: sV visible

        if (w < OUT_W) {
            float sx = 0.f, sy = 0.f, sxx = 0.f, syy = 0.f, sxy = 0.f;
#pragma unroll
            for (int j2 = 0; j2 < 7; ++j2) {
                sx  += sVx[w + j2];  sy  += sVy[w + j2];
                sxx += sVxx[w + j2]; syy += sVyy[w + j2];
                sxy += sVxy[w + j2];
            }
            float ux = sx * inv,  uy = sy * inv;
            float uxx = sxx * inv, uyy = syy * inv, uxy = sxy * inv;
            float vvx  = cn * (uxx - ux * ux);
            float vvy  = cn * (uyy - uy * uy);
            float vvxy = cn * (uxy - ux * uy);
            float A1 = 2.f * ux * uy + c1;
            float A2 = 2.f * vvxy + c2;
            float B1 = ux * ux + uy * uy + c1;
            float B2 = vvx + vvy + c2;
            lsum += (A1 * A2) / (B1 * B2);
        }
        __syncthreads();   // B: sV reads done before next iteration's writes

        if (hp < ROWS_OUT - 1) {
            // rows issued beyond the one we consume now (hp+7)
            const int rem = (ROWS_OUT - 2) - hp;
            if      (rem >= 3) WAIT_ASYNC(6);
            else if (rem == 2) WAIT_ASYNC(4);
            else if (rem == 1) WAIT_ASYNC(2);
            else               WAIT_ASYNC(0);
            const int jc = hp + 7;
            const int r  = hp % 7;
            float xn = stX[jc & 3][w], yn = stY[jc & 3][w];
            float xo = ringX[r][w],    yo = ringY[r][w];
            ringX[r][w] = xn; ringY[r][w] = yn;
            vx  += xn - xo;            vy  += yn - yo;
            vxx += xn * xn - xo * xo;  vyy += yn * yn - yo * yo;
            vxy += xn * yn - xo * yo;
        }
    }

    red[w] = lsum;
    __syncthreads();
    for (int s = 128; s; s >>= 1) {
        if (w < s) red[w] += red[w + s];
        __syncthreads();
    }
    if (w == 0) atomicAdd(acc, (double)red[0]);   // global_atomic_add_f64
}

__global__ __launch_bounds__(32)
void k_fin(const double* acc, float* out) {
    if (threadIdx.x == 0)
        out[0] = 1.0f - (float)(*acc * (1.0 / 20000000.0));  // F_*250*250
}

// ---------------------------------------------------------------------------
extern "C" void kernel_launch(void* const* d_in, const int* in_sizes, int n_in,
                              void* d_out, int out_size, void* d_ws, size_t ws_size,
                              hipStream_t stream) {
    (void)in_sizes; (void)n_in; (void)out_size; (void)ws_size;
    const float* X = (const float*)d_in[0];   // Xt [16,1,20,256,256] f32
    const float* Y = (const float*)d_in[1];   // Yt
    float*  out = (float*)d_out;              // scalar f32
    float*  dr  = (float*)d_ws;               // 20 per-t maxima
    double* acc = (double*)((char*)d_ws + 128);

    k_init<<<1, 32, 0, stream>>>(dr, acc);

    dim3 gmax(T_, B_, 8);
    k_framemax<<<gmax, 256, 0, stream>>>(Y, dr);

    dim3 gssim(F_, BANDS);
    k_ssim<<<gssim, 256, 0, stream>>>(X, Y, dr, acc);

    k_fin<<<1, 32, 0, stream>>>(acc, out);
}